// LightGCN_41704132444583
// MI455X (gfx1250) — compile-verified
//
#include <hip/hip_runtime.h>

typedef float v2f __attribute__((ext_vector_type(2)));
typedef float v8f __attribute__((ext_vector_type(8)));

#define DIM 64

// ---------------------------------------------------------------------------
// Kernel 1: E_cur = concat(user_emb, item_emb); acc = E_cur; E_next = 0
// One float4 per thread.
// ---------------------------------------------------------------------------
__global__ void lgcn_init(const float* __restrict__ ue, const float* __restrict__ ie,
                          float* __restrict__ Ecur, float* __restrict__ Enext,
                          float* __restrict__ acc, int u_elems, int total4) {
    int i = blockIdx.x * blockDim.x + threadIdx.x;     // float4 index
    if (i >= total4) return;
    int e4 = i * 4;
    float4 v = (e4 < u_elems) ? ((const float4*)ue)[i]
                              : ((const float4*)ie)[i - (u_elems >> 2)];
    ((float4*)Ecur)[i]  = v;
    ((float4*)acc)[i]   = v;
    ((float4*)Enext)[i] = make_float4(0.f, 0.f, 0.f, 0.f);
}

// ---------------------------------------------------------------------------
// Kernel 2: edge-parallel SpMM scatter. 16 threads per edge, one float4 each.
// Edge stream (rows/cols/vals) is streamed non-temporally so the 76.8 MB
// embedding table stays resident in the 192 MB L2; the fp32 atomic adds
// execute at L2 (global_atomic_add_f32, no return).
// ---------------------------------------------------------------------------
__global__ void lgcn_spmm(const int* __restrict__ rows, const int* __restrict__ cols,
                          const float* __restrict__ vals,
                          const float* __restrict__ Ein, float* __restrict__ Eout,
                          int n_edges) {
    long long gid = (long long)blockIdx.x * blockDim.x + threadIdx.x;
    long long e   = gid >> 4;
    if (e >= n_edges) return;
    int c4  = (int)(gid & 15);                         // which float4 of the row
    int col = __builtin_nontemporal_load(cols + e);
    int row = __builtin_nontemporal_load(rows + e);
    float val = __builtin_nontemporal_load(vals + e);
    float4 src = ((const float4*)(Ein + (long long)col * DIM))[c4];
    float* dst = Eout + (long long)row * DIM + c4 * 4;
    unsafeAtomicAdd(dst + 0, val * src.x);
    unsafeAtomicAdd(dst + 1, val * src.y);
    unsafeAtomicAdd(dst + 2, val * src.z);
    unsafeAtomicAdd(dst + 3, val * src.w);
}

// ---------------------------------------------------------------------------
// Kernel 3: acc += src; zbuf = 0  (zbuf becomes next layer's scatter target)
// ---------------------------------------------------------------------------
__global__ void lgcn_acc_zero(float* __restrict__ acc, const float* __restrict__ src,
                              float* __restrict__ zbuf, int total4) {
    int i = blockIdx.x * blockDim.x + threadIdx.x;
    if (i >= total4) return;
    float4 a = ((const float4*)acc)[i];
    float4 s = ((const float4*)src)[i];
    a.x += s.x; a.y += s.y; a.z += s.z; a.w += s.w;
    ((float4*)acc)[i]  = a;
    ((float4*)zbuf)[i] = make_float4(0.f, 0.f, 0.f, 0.f);
}

// ---------------------------------------------------------------------------
// Kernel 4: batched dot products via V_WMMA_F32_16X16X4_F32.
// One wave32 handles 16 (u,i) pairs: D = A(16x64) x B(64x16) as 16 chained
// k=4 WMMAs; diagonal D[m][m] = dot(u_m, i_m). Exact fp32 accumulation.
//
// A layout per ISA (32-bit A 16x4): lane L holds row M = L&15;
//   VGPR0 = K = 4*kk + 2*(L>=16), VGPR1 = K+1.  B mirrored (64x16 col-major).
// C/D layout (32-bit 16x16): N = lane&15, M = vgpr + 8*(lane>=16).
// ---------------------------------------------------------------------------
__global__ void lgcn_dot_wmma(const float* __restrict__ acc,
                              const int* __restrict__ U, const int* __restrict__ I,
                              float* __restrict__ out, int n_users, int batch) {
    int lane  = threadIdx.x & 31;
    int wave  = blockIdx.x * (blockDim.x >> 5) + (threadIdx.x >> 5);
    int pbase = wave * 16;
    if (pbase >= batch) return;          // whole wave exits together (EXEC stays all-1s)

    int m  = lane & 15;                  // row-in-tile for A, col-in-tile for B
    int hi = lane >> 4;

    long long urow = (long long)U[pbase + m] * DIM;
    long long irow = ((long long)n_users + (long long)I[pbase + m]) * DIM;

    v8f c = {};
#pragma unroll
    for (int kk = 0; kk < 16; ++kk) {
        int k0 = kk * 4 + hi * 2;
        v2f a = { acc[urow + k0], acc[urow + k0 + 1] };
        v2f b = { acc[irow + k0], acc[irow + k0 + 1] };
        c = __builtin_amdgcn_wmma_f32_16x16x4_f32(
                /*neg_a=*/false, a, /*neg_b=*/false, b,
                /*c_mod=*/(short)0, c, /*reuse_a=*/false, /*reuse_b=*/false);
    }

    // Extract diagonal D[M][M]; scale by (1/4)*(1/4) for out = acc/4 on both sides.
#pragma unroll
    for (int r = 0; r < 8; ++r) {
        int M = r + 8 * hi;
        if (M == m) out[pbase + M] = c[r] * 0.0625f;
    }
}

// ---------------------------------------------------------------------------
extern "C" void kernel_launch(void* const* d_in, const int* in_sizes, int n_in,
                              void* d_out, int out_size, void* d_ws, size_t ws_size,
                              hipStream_t stream) {
    const float* user_emb = (const float*)d_in[0];
    const float* item_emb = (const float*)d_in[1];
    const float* vals     = (const float*)d_in[2];
    const int*   rows     = (const int*)d_in[3];
    const int*   cols     = (const int*)d_in[4];
    const int*   U        = (const int*)d_in[5];
    const int*   I        = (const int*)d_in[6];
    float*       out      = (float*)d_out;

    const int n_users = in_sizes[0] / DIM;
    const int n_items = in_sizes[1] / DIM;
    const int n_edges = in_sizes[2];
    const int batch   = in_sizes[5];
    const int n       = n_users + n_items;

    const size_t n64    = (size_t)n * DIM;     // floats per buffer
    const int    total4 = (int)(n64 / 4);      // float4 count

    float* A   = (float*)d_ws;                 // E_cur
    float* B   = A + n64;                      // E_next (scatter target)
    float* acc = B + n64;                      // running sum

    const int ibl = 256;
    lgcn_init<<<(total4 + ibl - 1) / ibl, ibl, 0, stream>>>(
        user_emb, item_emb, A, B, acc, n_users * DIM, total4);

    float* cur = A;
    float* nxt = B;
    for (int layer = 0; layer < 3; ++layer) {
        long long threads = (long long)n_edges * 16;
        int blocks = (int)((threads + 255) / 256);
        lgcn_spmm<<<blocks, 256, 0, stream>>>(rows, cols, vals, cur, nxt, n_edges);
        lgcn_acc_zero<<<(total4 + ibl - 1) / ibl, ibl, 0, stream>>>(acc, nxt, cur, total4);
        float* t = cur; cur = nxt; nxt = t;    // freshly-zeroed buffer becomes next target
    }

    // 16 pairs per wave, 8 waves per block
    int waves  = batch / 16;
    int blocks = (waves + 7) / 8;
    lgcn_dot_wmma<<<blocks, 256, 0, stream>>>(acc, U, I, out, n_users, batch);
}